// LSTM_13606456393853
// MI455X (gfx1250) — compile-verified
//
#include <hip/hip_runtime.h>

#define B_SZ  64
#define T_SZ  2048
#define D_INP 128
#define H_SZ  512
#define NOUT  128
#define NWG   32
#define WROWS 64      // gate rows per WG (4 gates x 16 hidden units)
#define WPAD  520     // padded LDS row stride (floats) for weight tiles
#define GPAD  68      // padded LDS row stride (floats) for gates tile

typedef float v2f __attribute__((ext_vector_type(2)));
typedef float v8f __attribute__((ext_vector_type(8)));

__device__ __forceinline__ v8f wmma4(v2f a, v2f b, v8f c) {
  // D = A(16x4 f32) * B(4x16 f32) + C(16x16 f32)
  return __builtin_amdgcn_wmma_f32_16x16x4_f32(false, a, false, b, (short)0, c,
                                               false, false);
}

__device__ __forceinline__ float sigm(float x) {
  return 1.0f / (1.0f + __expf(-x));
}

// Monotonic grid barrier over NWG persistent workgroups.
__device__ __forceinline__ void gbarrier(unsigned* cnt, unsigned target) {
  __syncthreads();
  if (threadIdx.x == 0) {
    __threadfence();
    __hip_atomic_fetch_add(cnt, 1u, __ATOMIC_RELEASE, __HIP_MEMORY_SCOPE_AGENT);
    while (__hip_atomic_load(cnt, __ATOMIC_ACQUIRE, __HIP_MEMORY_SCOPE_AGENT) <
           target)
      __builtin_amdgcn_s_sleep(1);
  }
  __syncthreads();
}

// One K-phase of the per-step GEMM: A from global (row stride rsA floats),
// B from LDS (row stride WPAD). Each wave accumulates two 16x16 tiles
// (miBase, miBase+1) x (ni) using 4 independent WMMA chains.
__device__ __forceinline__ void gemm_phase(const float* __restrict__ Abase,
                                           int rsA, const float* Blds, int K,
                                           int miBase, int ni, int r, int hi,
                                           v8f& acc0, v8f& acc1, v8f& acc2,
                                           v8f& acc3) {
  const float* a0p = Abase + (size_t)(miBase * 16 + r) * rsA + 2 * hi;
  const float* a1p = a0p + (size_t)16 * rsA;
  const float* bp  = Blds + (ni * 16 + r) * WPAD + 2 * hi;
#pragma unroll 4
  for (int k = 0; k < K; k += 8) {
    v2f b0 = *(const v2f*)(bp + k);
    v2f b1 = *(const v2f*)(bp + k + 4);
    acc0 = wmma4(*(const v2f*)(a0p + k), b0, acc0);
    acc1 = wmma4(*(const v2f*)(a1p + k), b0, acc1);
    acc2 = wmma4(*(const v2f*)(a0p + k + 4), b1, acc2);
    acc3 = wmma4(*(const v2f*)(a1p + k + 4), b1, acc3);
  }
}

__global__ void __launch_bounds__(256, 1)
lstm_persist(const float* __restrict__ x_tr,
             const float* __restrict__ Wih0, const float* __restrict__ Whh0,
             const float* __restrict__ bi0, const float* __restrict__ bh0,
             const float* __restrict__ Wih1, const float* __restrict__ Whh1,
             const float* __restrict__ bi1, const float* __restrict__ bh1,
             const float* __restrict__ fcw, const float* __restrict__ fcb,
             float* __restrict__ h0seq, float* __restrict__ ping,
             float* __restrict__ out, unsigned* __restrict__ bar) {
  extern __shared__ float smem[];
  float* WHH   = smem;                    // 64 x 520
  float* WIH   = WHH + WROWS * WPAD;      // 64 x 520
  float* GATES = WIH + WROWS * WPAD;      // 64 x 68
  float* BIAS  = GATES + B_SZ * GPAD;     // 64

  const int b = blockIdx.x;               // 0..31: owns hidden units [16b,16b+16)
  const int tid = threadIdx.x;
  const int wave = tid >> 5, lane = tid & 31;
  const int r = lane & 15, hi = lane >> 4;
  const int ni = wave & 3;                // gate tile: 0=i 1=f 2=g 3=o
  const int miBase = (wave >> 2) << 1;    // batch tiles {0,1} or {2,3}
  unsigned epoch = 0;
  float c_reg[4];

  for (int layer = 0; layer < 2; ++layer) {
    const int D = layer ? H_SZ : D_INP;
    const float* Wih = layer ? Wih1 : Wih0;
    const float* Whh = layer ? Whh1 : Whh0;
    const float* bi = layer ? bi1 : bi0;
    const float* bh = layer ? bh1 : bh0;

    // Stage this WG's weight slices into LDS (once per layer).
    for (int idx = tid; idx < WROWS * H_SZ; idx += 256) {
      int n = idx >> 9, k = idx & (H_SZ - 1);
      int grow = ((n >> 4) << 9) + (b << 4) + (n & 15);  // gate*512 + 16b + hu
      WHH[n * WPAD + k] = Whh[(size_t)grow * H_SZ + k];
    }
    for (int idx = tid; idx < WROWS * D; idx += 256) {
      int n = idx / D, k = idx % D;
      int grow = ((n >> 4) << 9) + (b << 4) + (n & 15);
      WIH[n * WPAD + k] = Wih[(size_t)grow * D + k];
    }
    for (int n = tid; n < WROWS; n += 256) {
      int grow = ((n >> 4) << 9) + (b << 4) + (n & 15);
      BIAS[n] = bi[grow] + bh[grow];
    }
#pragma unroll
    for (int j = 0; j < 4; ++j) c_reg[j] = 0.0f;
    __syncthreads();

    for (int t = 0; t < T_SZ; ++t) {
      v8f acc0 = {}, acc1 = {}, acc2 = {}, acc3 = {};
      if (t != 0) {  // h_{t-1} @ W_hh^T  (skip at t=0: h0 == 0)
        const float* Ah = layer ? (ping + (size_t)((t - 1) & 1) * B_SZ * H_SZ)
                                : (h0seq + (size_t)(t - 1) * B_SZ * H_SZ);
        gemm_phase(Ah, H_SZ, WHH, H_SZ, miBase, ni, r, hi, acc0, acc1, acc2,
                   acc3);
      }
      // x_t @ W_ih^T (layer1's x_t is layer0's h sequence — fused, never
      // materialized as a [B,T,4H] tensor)
      const float* Ax = layer ? (h0seq + (size_t)t * B_SZ * H_SZ)
                              : (x_tr + (size_t)t * B_SZ * D_INP);
      gemm_phase(Ax, D, WIH, D, miBase, ni, r, hi, acc0, acc1, acc2, acc3);
      acc0 += acc2;
      acc1 += acc3;

      // Spill gate pre-activations to LDS (C layout: col=lane&15, row=vr+8*hi)
#pragma unroll
      for (int vr = 0; vr < 8; ++vr) {
        GATES[(miBase * 16 + vr + 8 * hi) * GPAD + ni * 16 + r] = acc0[vr];
        GATES[((miBase + 1) * 16 + vr + 8 * hi) * GPAD + ni * 16 + r] = acc1[vr];
      }
      __syncthreads();

      float* hdst = layer ? (ping + (size_t)(t & 1) * B_SZ * H_SZ)
                          : (h0seq + (size_t)t * B_SZ * H_SZ);
#pragma unroll
      for (int j = 0; j < 4; ++j) {
        int e = tid + j * 256;  // (batch m, hidden hu) pair, fixed per thread
        int m = e >> 4, hu = e & 15;
        float gi = GATES[m * GPAD + hu] + BIAS[hu];
        float gf = GATES[m * GPAD + 16 + hu] + BIAS[16 + hu];
        float gg = GATES[m * GPAD + 32 + hu] + BIAS[32 + hu];
        float go = GATES[m * GPAD + 48 + hu] + BIAS[48 + hu];
        float iv = sigm(gi), fv = sigm(gf);
        float gv = tanhf(gg), ov = sigm(go);
        float c = fv * c_reg[j] + iv * gv;
        c_reg[j] = c;
        hdst[(size_t)m * H_SZ + (b << 4) + hu] = ov * tanhf(c);
      }
      gbarrier(bar, (++epoch) * NWG);
    }
  }

  // FC on last timestep's h1 (ping[(T-1)&1]): out[64,128] = h @ fcw^T + fcb
  if (b < 8 && wave < 4) {
    const float* hfin = ping + (size_t)((T_SZ - 1) & 1) * B_SZ * H_SZ;
    int mi = wave;
    v8f acc = {}, accB = {};
    const float* ap = hfin + (size_t)(mi * 16 + r) * H_SZ + 2 * hi;
    const float* bpg = fcw + (size_t)((b << 4) + r) * H_SZ + 2 * hi;
#pragma unroll 4
    for (int k = 0; k < H_SZ; k += 8) {
      acc = wmma4(*(const v2f*)(ap + k), *(const v2f*)(bpg + k), acc);
      accB = wmma4(*(const v2f*)(ap + k + 4), *(const v2f*)(bpg + k + 4), accB);
    }
    acc += accB;
    float bias = fcb[(b << 4) + r];
#pragma unroll
    for (int vr = 0; vr < 8; ++vr)
      out[(size_t)(mi * 16 + vr + 8 * hi) * NOUT + (b << 4) + r] =
          acc[vr] + bias;
  }
}

// [B][T][D] -> [T][B][D] so the recurrent A-operand rows are 512B apart.
__global__ void xpose_kernel(const float* __restrict__ x,
                             float* __restrict__ x_tr) {
  int t = blockIdx.x & (T_SZ - 1);
  int m = blockIdx.x >> 11;
  int k = threadIdx.x;
  x_tr[((size_t)t * B_SZ + m) * D_INP + k] = x[((size_t)m * T_SZ + t) * D_INP + k];
}

extern "C" void kernel_launch(void* const* d_in, const int* in_sizes, int n_in,
                              void* d_out, int out_size, void* d_ws,
                              size_t ws_size, hipStream_t stream) {
  (void)in_sizes; (void)n_in; (void)out_size; (void)ws_size;
  const float* x    = (const float*)d_in[0];
  const float* Wih0 = (const float*)d_in[1];
  const float* Whh0 = (const float*)d_in[2];
  const float* bi0  = (const float*)d_in[3];
  const float* bh0  = (const float*)d_in[4];
  const float* Wih1 = (const float*)d_in[5];
  const float* Whh1 = (const float*)d_in[6];
  const float* bi1  = (const float*)d_in[7];
  const float* bh1  = (const float*)d_in[8];
  const float* fcw  = (const float*)d_in[9];
  const float* fcb  = (const float*)d_in[10];
  float* out = (float*)d_out;

  char* ws = (char*)d_ws;
  unsigned* bar = (unsigned*)ws;                                   // 256 B
  float* ping = (float*)(ws + 256);                                // 256 KB
  float* x_tr = (float*)(ws + 256 + (size_t)2 * B_SZ * H_SZ * 4);  // 64 MB
  float* h0seq = (float*)(ws + 256 + (size_t)2 * B_SZ * H_SZ * 4 +
                          (size_t)B_SZ * T_SZ * D_INP * 4);        // 256 MB

  hipMemsetAsync(bar, 0, 256, stream);
  xpose_kernel<<<B_SZ * T_SZ, D_INP, 0, stream>>>(x, x_tr);

  size_t shmem =
      (size_t)(2 * WROWS * WPAD + B_SZ * GPAD + WROWS) * sizeof(float);
  hipFuncSetAttribute((const void*)lstm_persist,
                      hipFuncAttributeMaxDynamicSharedMemorySize, (int)shmem);
  lstm_persist<<<NWG, 256, shmem, stream>>>(x_tr, Wih0, Whh0, bi0, bh0, Wih1,
                                            Whh1, bi1, bh1, fcw, fcb, h0seq,
                                            ping, out, bar);
}